// SOMKeras_87144886436418
// MI455X (gfx1250) — compile-verified
//
#include <hip/hip_runtime.h>
#include <math.h>

#define BATCH 256
#define NNEUR 4096
#define DIMK  512
#define NOUT  (NNEUR*DIMK)

typedef float v2f __attribute__((ext_vector_type(2)));
typedef float v8f __attribute__((ext_vector_type(8)));

// workspace layout (byte offsets)
#define WS_KEYS    0u          // 256 * u64                = 2048 B
#define WS_SCAL    2048u       // [0]=1/(2*sigma^2) [1]=taxa/B
#define WS_W2      4096u       // 4096 f32                 = 16 KB
#define WS_HSUM    20480u      // 4096 f32                 = 16 KB
#define WS_XT      36864u      // 512*256 f32              = 512 KB
#define WS_HT      561152u     // 4096*256 f32             = 4 MB

// order-preserving float -> uint map (monotone for all signs)
__device__ __forceinline__ unsigned int ordered_bits(float f) {
  unsigned int u = __float_as_uint(f);
  return (u & 0x80000000u) ? ~u : (u | 0x80000000u);
}

__global__ void k_prep(const int* __restrict__ passo, const int* __restrict__ total,
                       float* __restrict__ out, unsigned long long* __restrict__ keys,
                       float* __restrict__ scal) {
  int t = threadIdx.x;
  keys[t] = 0xFFFFFFFFFFFFFFFFull;
  if (t == 0) {
    float decay = expf(-(float)passo[0] / (float)total[0]);
    float taxa  = 0.5f  * decay;
    float sigma = 32.0f * decay;
    out[NOUT]     = taxa;
    out[NOUT + 1] = sigma;
    scal[0] = 1.0f / (2.0f * sigma * sigma);
    scal[1] = taxa / (float)BATCH;
  }
}

// per-neuron squared norm: one wave per row
__global__ __launch_bounds__(256) void k_w2(const float* __restrict__ pesos,
                                            float* __restrict__ w2) {
  int lane = threadIdx.x & 31, wid = threadIdx.x >> 5;
  int row  = blockIdx.x * 8 + wid;
  const float4* r = (const float4*)(pesos + (size_t)row * DIMK);
  float s = 0.f;
#pragma unroll
  for (int i = 0; i < 4; ++i) {
    float4 v = r[lane + 32 * i];
    s += v.x * v.x + v.y * v.y + v.z * v.z + v.w * v.w;
  }
#pragma unroll
  for (int off = 16; off >= 1; off >>= 1) s += __shfl_xor(s, off, 32);
  if (lane == 0) w2[row] = s;
}

// X (B x D) -> XT (D x B), coalesced writes
__global__ __launch_bounds__(256) void k_xt(const float* __restrict__ X,
                                            float* __restrict__ XT) {
  int d = blockIdx.x;   // 0..511
  int b = threadIdx.x;  // 0..255
  XT[(size_t)d * BATCH + b] = X[(size_t)b * DIMK + d];
}

// GEMM1: S = X * W^T via v_wmma_f32_16x16x4_f32, software-pipelined,
// fused argmin via u64 atomic-min keys
__global__ __launch_bounds__(256) void k_dist_argmin(
    const float* __restrict__ X, const float* __restrict__ pesos,
    const float* __restrict__ w2, unsigned long long* __restrict__ keys) {
  __shared__ float xs[16][516];              // pitch 516 -> conflict-free
  const int t = threadIdx.x;
  const int lane = t & 31, wid = t >> 5;
  const int m_base = blockIdx.y * 16;        // batch tile
  const int n_wave = blockIdx.x * 512 + wid * 64;

  { // stage 16x512 X tile into LDS (8 float4 per thread)
    const float4* src = (const float4*)(X + (size_t)m_base * DIMK);
#pragma unroll
    for (int i = 0; i < 8; ++i) {
      int p = t + i * 256;                   // float4 index, 128 per row
      int row = p >> 7, c4 = p & 127;
      float4 v = src[(size_t)row * 128 + c4];
      *(float4*)&xs[row][c4 * 4] = v;
    }
  }
  __syncthreads();

  const int lh = lane >> 4, mr = lane & 15;
  const float* wr0 = pesos + (size_t)(n_wave +  0 + mr) * DIMK;
  const float* wr1 = pesos + (size_t)(n_wave + 16 + mr) * DIMK;
  const float* wr2 = pesos + (size_t)(n_wave + 32 + mr) * DIMK;
  const float* wr3 = pesos + (size_t)(n_wave + 48 + mr) * DIMK;

  v8f acc[4] = {};

  // ---- software-pipelined K loop (prefetch depth 1, peeled tail) ----
  int ko = 2 * lh;
  v2f a  = *(const v2f*)&xs[mr][ko];
  v2f b0 = *(const v2f*)(wr0 + ko);
  v2f b1 = *(const v2f*)(wr1 + ko);
  v2f b2 = *(const v2f*)(wr2 + ko);
  v2f b3 = *(const v2f*)(wr3 + ko);

#pragma unroll 2
  for (int k0 = 0; k0 < DIMK - 4; k0 += 4) {
    const int kn = k0 + 4 + 2 * lh;
    // issue next iteration's loads first (overlap with WMMAs below)
    v2f a_n  = *(const v2f*)&xs[mr][kn];
    v2f b0_n = *(const v2f*)(wr0 + kn);
    v2f b1_n = *(const v2f*)(wr1 + kn);
    v2f b2_n = *(const v2f*)(wr2 + kn);
    v2f b3_n = *(const v2f*)(wr3 + kn);

    acc[0] = __builtin_amdgcn_wmma_f32_16x16x4_f32(false, a, false, b0, (short)0, acc[0], false, false);
    acc[1] = __builtin_amdgcn_wmma_f32_16x16x4_f32(false, a, false, b1, (short)0, acc[1], false, false);
    acc[2] = __builtin_amdgcn_wmma_f32_16x16x4_f32(false, a, false, b2, (short)0, acc[2], false, false);
    acc[3] = __builtin_amdgcn_wmma_f32_16x16x4_f32(false, a, false, b3, (short)0, acc[3], false, false);

    a = a_n; b0 = b0_n; b1 = b1_n; b2 = b2_n; b3 = b3_n;
  }
  // peeled final iteration
  acc[0] = __builtin_amdgcn_wmma_f32_16x16x4_f32(false, a, false, b0, (short)0, acc[0], false, false);
  acc[1] = __builtin_amdgcn_wmma_f32_16x16x4_f32(false, a, false, b1, (short)0, acc[1], false, false);
  acc[2] = __builtin_amdgcn_wmma_f32_16x16x4_f32(false, a, false, b2, (short)0, acc[2], false, false);
  acc[3] = __builtin_amdgcn_wmma_f32_16x16x4_f32(false, a, false, b3, (short)0, acc[3], false, false);

  // epilogue: val = w2[n] - 2*S  (x^2 constant per row, drops out of argmin)
#pragma unroll
  for (int j = 0; j < 4; ++j) {
    int n = n_wave + j * 16 + mr;
    float w2n = w2[n];
#pragma unroll
    for (int i = 0; i < 8; ++i) {
      float val = w2n - 2.0f * acc[j][i];
      unsigned long long key =
          ((unsigned long long)ordered_bits(val) << 32) | (unsigned int)n;
#pragma unroll
      for (int off = 8; off >= 1; off >>= 1) {
        unsigned long long o = __shfl_xor(key, off, 32);
        key = (o < key) ? o : key;
      }
      if (mr == 0) {
        int m = m_base + i + 8 * lh;         // C/D layout: VGPR i -> row i / i+8
        atomicMin(&keys[m], key);
      }
    }
  }
}

// h[b,n] transposed to hT[n][b]; block-reduced h_sum[n]
__global__ __launch_bounds__(256) void k_h(
    const float* __restrict__ loc, const unsigned long long* __restrict__ keys,
    const float* __restrict__ scal, float* __restrict__ hT, float* __restrict__ h_sum) {
  int n = blockIdx.x, b = threadIdx.x;
  float inv2s2 = scal[0];
  unsigned int idx = (unsigned int)(keys[b] & 0xFFFFFFFFull);
  float dx = loc[2 * n]     - loc[2 * idx];
  float dy = loc[2 * n + 1] - loc[2 * idx + 1];
  float h = expf(-(dx * dx + dy * dy) * inv2s2);
  hT[(size_t)n * BATCH + b] = h;

  float s = h;
#pragma unroll
  for (int off = 16; off >= 1; off >>= 1) s += __shfl_xor(s, off, 32);
  __shared__ float red[8];
  if ((b & 31) == 0) red[b >> 5] = s;
  __syncthreads();
  if (b == 0) {
    float tot = 0.f;
#pragma unroll
    for (int i = 0; i < 8; ++i) tot += red[i];
    h_sum[n] = tot;
  }
}

// GEMM2: hX = h^T * X via WMMA (software-pipelined), fused SOM weight update
__global__ __launch_bounds__(256) void k_update(
    const float* __restrict__ hT, const float* __restrict__ XT,
    const float* __restrict__ pesos, const float* __restrict__ h_sum,
    const float* __restrict__ scal, float* __restrict__ out) {
  __shared__ float hs[16][260];
  const int t = threadIdx.x;
  const int lane = t & 31, wid = t >> 5;
  const int n_base = blockIdx.x * 16;

  { // stage 16x256 hT tile (4 float4 per thread)
    const float4* src = (const float4*)(hT + (size_t)n_base * BATCH);
#pragma unroll
    for (int i = 0; i < 4; ++i) {
      int p = t + i * 256;                   // float4 index, 64 per row
      int row = p >> 6, c4 = p & 63;
      float4 v = src[(size_t)row * 64 + c4];
      *(float4*)&hs[row][c4 * 4] = v;
    }
  }
  __syncthreads();

  const int lh = lane >> 4, mr = lane & 15;
  const int d0 = wid * 64;
  const float* xt0 = XT + (size_t)(d0 +  0 + mr) * BATCH;
  const float* xt1 = XT + (size_t)(d0 + 16 + mr) * BATCH;
  const float* xt2 = XT + (size_t)(d0 + 32 + mr) * BATCH;
  const float* xt3 = XT + (size_t)(d0 + 48 + mr) * BATCH;

  v8f acc[4] = {};

  // ---- software-pipelined K loop (prefetch depth 1, peeled tail) ----
  int ko = 2 * lh;
  v2f a  = *(const v2f*)&hs[mr][ko];
  v2f b0 = *(const v2f*)(xt0 + ko);
  v2f b1 = *(const v2f*)(xt1 + ko);
  v2f b2 = *(const v2f*)(xt2 + ko);
  v2f b3 = *(const v2f*)(xt3 + ko);

#pragma unroll 2
  for (int k0 = 0; k0 < BATCH - 4; k0 += 4) {
    const int kn = k0 + 4 + 2 * lh;
    v2f a_n  = *(const v2f*)&hs[mr][kn];
    v2f b0_n = *(const v2f*)(xt0 + kn);
    v2f b1_n = *(const v2f*)(xt1 + kn);
    v2f b2_n = *(const v2f*)(xt2 + kn);
    v2f b3_n = *(const v2f*)(xt3 + kn);

    acc[0] = __builtin_amdgcn_wmma_f32_16x16x4_f32(false, a, false, b0, (short)0, acc[0], false, false);
    acc[1] = __builtin_amdgcn_wmma_f32_16x16x4_f32(false, a, false, b1, (short)0, acc[1], false, false);
    acc[2] = __builtin_amdgcn_wmma_f32_16x16x4_f32(false, a, false, b2, (short)0, acc[2], false, false);
    acc[3] = __builtin_amdgcn_wmma_f32_16x16x4_f32(false, a, false, b3, (short)0, acc[3], false, false);

    a = a_n; b0 = b0_n; b1 = b1_n; b2 = b2_n; b3 = b3_n;
  }
  acc[0] = __builtin_amdgcn_wmma_f32_16x16x4_f32(false, a, false, b0, (short)0, acc[0], false, false);
  acc[1] = __builtin_amdgcn_wmma_f32_16x16x4_f32(false, a, false, b1, (short)0, acc[1], false, false);
  acc[2] = __builtin_amdgcn_wmma_f32_16x16x4_f32(false, a, false, b2, (short)0, acc[2], false, false);
  acc[3] = __builtin_amdgcn_wmma_f32_16x16x4_f32(false, a, false, b3, (short)0, acc[3], false, false);

  float scale = scal[1];
  float hsum_i[8];
#pragma unroll
  for (int i = 0; i < 8; ++i) hsum_i[i] = h_sum[n_base + i + 8 * lh];

#pragma unroll
  for (int j = 0; j < 4; ++j) {
    int d = d0 + j * 16 + mr;
#pragma unroll
    for (int i = 0; i < 8; ++i) {
      int n = n_base + i + 8 * lh;
      size_t off = (size_t)n * DIMK + d;
      float p = pesos[off];
      out[off] = p + scale * (acc[j][i] - hsum_i[i] * p);
    }
  }
}

extern "C" void kernel_launch(void* const* d_in, const int* in_sizes, int n_in,
                              void* d_out, int out_size, void* d_ws, size_t ws_size,
                              hipStream_t stream) {
  (void)in_sizes; (void)n_in; (void)out_size; (void)ws_size;
  const float* X     = (const float*)d_in[0];
  const float* pesos = (const float*)d_in[1];
  const float* loc   = (const float*)d_in[2];
  const int*   passo = (const int*)d_in[3];
  const int*   total = (const int*)d_in[4];
  float* out = (float*)d_out;

  char* ws = (char*)d_ws;
  unsigned long long* keys = (unsigned long long*)(ws + WS_KEYS);
  float* scal = (float*)(ws + WS_SCAL);
  float* w2   = (float*)(ws + WS_W2);
  float* hsum = (float*)(ws + WS_HSUM);
  float* XT   = (float*)(ws + WS_XT);
  float* hT   = (float*)(ws + WS_HT);

  k_prep<<<1, 256, 0, stream>>>(passo, total, out, keys, scal);
  k_w2<<<NNEUR / 8, 256, 0, stream>>>(pesos, w2);
  k_xt<<<DIMK, 256, 0, stream>>>(X, XT);
  k_dist_argmin<<<dim3(NNEUR / 512, BATCH / 16), 256, 0, stream>>>(X, pesos, w2, keys);
  k_h<<<NNEUR, 256, 0, stream>>>(loc, keys, scal, hT, hsum);
  k_update<<<NNEUR / 16, 256, 0, stream>>>(hT, XT, pesos, hsum, scal, out);
}